// LowrankLearnableHash_46454366273928
// MI455X (gfx1250) — compile-verified
//
#include <hip/hip_runtime.h>
#include <cstdint>

// ---------------------------------------------------------------------------
// LowrankLearnableHash.compute_features for MI455X (gfx1250, wave32)
//
// Strategy:
//  1. Transpose plane0/1/2 [24,512,512] -> [512*512, 24] and
//     features [32,128,128,128] -> [128^3, 32] into d_ws, staging through LDS
//     with CDNA5 async global->LDS copies (ASYNCcnt, non-temporal reads) so
//     each bilinear/trilinear corner becomes a contiguous 96B/128B record.
//  2. Main pass: one wave32 per point; lane == channel. Every plane corner is
//     one coalesced 96B wave load, every feature corner one coalesced 128B
//     wave load; rank-8 reduction via shfl_xor butterflies. Output is stored
//     non-temporally so the 128MB result stream does not evict the 256MB
//     feature volume from the 192MB L2.
// ---------------------------------------------------------------------------

#define PLANE_RES 512
#define FEAT_RES  128
#define PLANE_C   24   // OUT_DIM * RANK
#define FEAT_C    32
#define RANK      8

__device__ __forceinline__ uint32_t lds_off_of(const void* p) {
  // addrspace(3) objects live at low 32-bit offsets of the LDS aperture;
  // truncating the flat address yields the wave-relative LDS byte offset.
  return (uint32_t)(uint64_t)(uintptr_t)p;
}

__device__ __forceinline__ void async_global_to_lds_b32_nt(uint32_t lds_byte_off,
                                                           const void* gaddr) {
  // CDNA5 async copy: one dword from global memory into LDS, tracked with ASYNCcnt.
  // TH_LOAD_NT: source is streamed exactly once; don't pollute caches.
  asm volatile("global_load_async_to_lds_b32 %0, %1, off th:TH_LOAD_NT"
               :: "v"(lds_byte_off), "v"(gaddr)
               : "memory");
}

__device__ __forceinline__ void wait_asynccnt0() {
  asm volatile("s_wait_asynccnt 0" ::: "memory");
}

// Transpose [C, nPos] -> [nPos, C]. 256 positions per block, staged via LDS
// using async global->LDS loads; +1 padding on the LDS row stride kills the
// 64-bank conflicts in the shuffled read-out phase.
template <int C>
__global__ void __launch_bounds__(256)
transpose_ch_last(const float* __restrict__ in, float* __restrict__ out,
                  long nPos) {
  constexpr int POS = 256;
  constexpr int STRIDE = POS + 1;
  __shared__ float tile[C * STRIDE];
  const int t = threadIdx.x;
  const long base = (long)blockIdx.x * POS;
  const uint32_t tbase = lds_off_of(&tile[0]);

#pragma unroll
  for (int c = 0; c < C; ++c) {
    // coalesced 128B/wave read from global, async write into LDS
    async_global_to_lds_b32_nt(tbase + (uint32_t)(c * STRIDE + t) * 4u,
                               in + (long)c * nPos + base + t);
  }
  wait_asynccnt0();   // each wave drains its own ASYNCcnt ...
  __syncthreads();    // ... then the barrier makes all waves' data visible

#pragma unroll
  for (int i = 0; i < C; ++i) {
    const int o = i * POS + t;      // 0 .. C*256-1, consecutive across lanes
    const int p = o / C;
    const int c = o - p * C;
    out[(base + p) * C + c] = tile[c * STRIDE + p];  // coalesced store
  }
}

// Main pass: one wave per point. Strides parameterize both the transposed
// (channel-last) fast path and the original-layout fallback.
__global__ void __launch_bounds__(256)
lowrank_hash_kernel(const float* __restrict__ pts,
                    const float* __restrict__ pl0,
                    const float* __restrict__ pl1,
                    const float* __restrict__ pl2,
                    const float* __restrict__ feat,
                    float* __restrict__ out, int n_pts,
                    long planePosStride, long planeChStride,
                    long featPosStride, long featChStride) {
  const int lane = threadIdx.x & 31;
  const int wave = threadIdx.x >> 5;
  int n = blockIdx.x * (blockDim.x >> 5) + wave;
  if (n >= n_pts) return;              // wave-uniform exit
  n = __builtin_amdgcn_readfirstlane(n);  // scalarize -> SMEM point loads

  const float px = pts[3 * n + 0];
  const float py = pts[3 * n + 1];
  const float pz = pts[3 * n + 2];

  const int cc = lane < PLANE_C ? lane : 0;  // clamp so loads stay in-bounds
  const float* plane[3] = {pl0, pl1, pl2};
  // combs: plane0 <- (p0,p1), plane1 <- (p0,p2), plane2 <- (p1,p2)
  // grid_sample: column 0 -> W (fast axis), column 1 -> H.
  const float gx[3] = {px, px, py};
  const float gy[3] = {py, pz, pz};

  float prod = 1.0f;  // per-lane channel product across the three planes
#pragma unroll
  for (int pl = 0; pl < 3; ++pl) {
    const float ux = (gx[pl] + 1.0f) * 0.5f * (float)(PLANE_RES - 1);
    const float uy = (gy[pl] + 1.0f) * 0.5f * (float)(PLANE_RES - 1);
    const float lxf = floorf(ux), lyf = floorf(uy);
    const float fx = ux - lxf, fy = uy - lyf;
    const int lx = (int)lxf, ly = (int)lyf;
    float s = 0.0f;
#pragma unroll
    for (int dy = 0; dy < 2; ++dy) {
#pragma unroll
      for (int dx = 0; dx < 2; ++dx) {
        const int ix = lx + dx, iy = ly + dy;
        const bool valid = ((unsigned)ix < (unsigned)PLANE_RES) &
                           ((unsigned)iy < (unsigned)PLANE_RES);
        float w = (dx ? fx : 1.0f - fx) * (dy ? fy : 1.0f - fy);
        w = valid ? w : 0.0f;
        const int icx = min(max(ix, 0), PLANE_RES - 1);
        const int icy = min(max(iy, 0), PLANE_RES - 1);
        const long pos = (long)(icy * PLANE_RES + icx);
        const float v = plane[pl][pos * planePosStride + (long)cc * planeChStride];
        s = fmaf(w, v, s);
      }
    }
    prod *= s;
  }

  // Rank reduction: channel c = dim*8 + r; sum r within 8-lane groups.
  float sum = prod;
  sum += __shfl_xor(sum, 1, 32);
  sum += __shfl_xor(sum, 2, 32);
  sum += __shfl_xor(sum, 4, 32);
  const float i0 = __shfl(sum, 0, 32);   // channels 0-7   -> x (W)
  const float i1 = __shfl(sum, 8, 32);   // channels 8-15  -> y (H)
  const float i2 = __shfl(sum, 16, 32);  // channels 16-23 -> z (D)

  // Trilinear lookup in the feature volume; lane == output channel.
  const float ux = (i0 + 1.0f) * 0.5f * (float)(FEAT_RES - 1);
  const float uy = (i1 + 1.0f) * 0.5f * (float)(FEAT_RES - 1);
  const float uz = (i2 + 1.0f) * 0.5f * (float)(FEAT_RES - 1);
  const float lxf = floorf(ux), lyf = floorf(uy), lzf = floorf(uz);
  const float fx = ux - lxf, fy = uy - lyf, fz = uz - lzf;
  const int lx = (int)lxf, ly = (int)lyf, lz = (int)lzf;

  float acc = 0.0f;
#pragma unroll
  for (int dz = 0; dz < 2; ++dz) {
#pragma unroll
    for (int dy = 0; dy < 2; ++dy) {
#pragma unroll
      for (int dx = 0; dx < 2; ++dx) {
        const int ix = lx + dx, iy = ly + dy, iz = lz + dz;
        const bool valid = ((unsigned)ix < (unsigned)FEAT_RES) &
                           ((unsigned)iy < (unsigned)FEAT_RES) &
                           ((unsigned)iz < (unsigned)FEAT_RES);
        float w = (dx ? fx : 1.0f - fx) * (dy ? fy : 1.0f - fy) *
                  (dz ? fz : 1.0f - fz);
        w = valid ? w : 0.0f;
        const int icx = min(max(ix, 0), FEAT_RES - 1);
        const int icy = min(max(iy, 0), FEAT_RES - 1);
        const int icz = min(max(iz, 0), FEAT_RES - 1);
        const long pos = ((long)icz * FEAT_RES + icy) * FEAT_RES + icx;
        acc = fmaf(w, feat[pos * featPosStride + (long)lane * featChStride], acc);
      }
    }
  }
  // Never re-read: keep this 128MB stream out of L2 (preserve feature lines).
  __builtin_nontemporal_store(acc, &out[(long)n * FEAT_C + lane]);
}

extern "C" void kernel_launch(void* const* d_in, const int* in_sizes, int n_in,
                              void* d_out, int out_size, void* d_ws,
                              size_t ws_size, hipStream_t stream) {
  const float* pts  = (const float*)d_in[0];
  const float* pl0  = (const float*)d_in[1];
  const float* pl1  = (const float*)d_in[2];
  const float* pl2  = (const float*)d_in[3];
  const float* feat = (const float*)d_in[4];
  float* out = (float*)d_out;

  const int n = in_sizes[0] / 3;  // 1,000,000
  const long planePos = (long)PLANE_RES * PLANE_RES;          // 262,144
  const long featPos  = (long)FEAT_RES * FEAT_RES * FEAT_RES; // 2,097,152
  const size_t planeElems = (size_t)PLANE_C * planePos;
  const size_t featElems  = (size_t)FEAT_C * featPos;
  const size_t need = (3 * planeElems + featElems) * sizeof(float); // ~328 MB

  const int waves_per_block = 8;  // 256 threads = 8 wave32
  const int grid = (n + waves_per_block - 1) / waves_per_block;

  if (ws_size >= need) {
    float* p0t = (float*)d_ws;
    float* p1t = p0t + planeElems;
    float* p2t = p1t + planeElems;
    float* ft  = p2t + planeElems;

    const int pBlocks = (int)(planePos / 256);  // 1024
    const int fBlocks = (int)(featPos / 256);   // 8192
    transpose_ch_last<PLANE_C><<<pBlocks, 256, 0, stream>>>(pl0, p0t, planePos);
    transpose_ch_last<PLANE_C><<<pBlocks, 256, 0, stream>>>(pl1, p1t, planePos);
    transpose_ch_last<PLANE_C><<<pBlocks, 256, 0, stream>>>(pl2, p2t, planePos);
    transpose_ch_last<FEAT_C><<<fBlocks, 256, 0, stream>>>(feat, ft, featPos);

    lowrank_hash_kernel<<<grid, 256, 0, stream>>>(
        pts, p0t, p1t, p2t, ft, out, n,
        /*planePosStride=*/PLANE_C, /*planeChStride=*/1,
        /*featPosStride=*/FEAT_C,  /*featChStride=*/1);
  } else {
    // Fallback: original [C, spatial] layout (strided channel gathers).
    lowrank_hash_kernel<<<grid, 256, 0, stream>>>(
        pts, pl0, pl1, pl2, feat, out, n,
        /*planePosStride=*/1, /*planeChStride=*/planePos,
        /*featPosStride=*/1,  /*featChStride=*/featPos);
  }
}